// KinematicChainLoss_32341103739417
// MI455X (gfx1250) — compile-verified
//
#include <hip/hip_runtime.h>

// KinematicChainLoss for MI455X (gfx1250, wave32).
// B=256, I=J=256 (S*M = T*N), output = scalar mean over (B,S,T).
// Pairwise dot products -> V_WMMA_F32_16X16X4_F32 (f32 matrix pipe);
// transcendental tail (sqrt + acos) on VALU with raw v_sqrt_f32 and an
// A&S 4.4.45 acos approximation (|err| <= 2e-8, inputs pre-clamped).

typedef __attribute__((ext_vector_type(2))) float v2f;
typedef __attribute__((ext_vector_type(8))) float v8f;

#define IJ     256   // S*M = T*N
#define POS_K  8     // [px,1,|px|^2,0..] dot [-2py,|py|^2,1,0..] = d2
#define ORI_K  12    // 9 rotation elements + zero pad
#define EPSF   1e-7f
#define INV_PI 0.31830988618379067154f
#define PI_F   3.14159265358979323846f

__device__ __forceinline__ float fast_sqrtf(float x) {
    return __builtin_amdgcn_sqrtf(x);        // raw v_sqrt_f32, ~1 ulp
}

// acos for x in [-1+eps, 1-eps]; Abramowitz-Stegun 4.4.45 (|err| <= 2e-8).
__device__ __forceinline__ float fast_acosf(float x) {
    float ax = __builtin_fabsf(x);
    float p =              -0.0012624911f;
    p = __builtin_fmaf(p, ax, 0.0066700901f);
    p = __builtin_fmaf(p, ax, -0.0170881256f);
    p = __builtin_fmaf(p, ax, 0.0308918810f);
    p = __builtin_fmaf(p, ax, -0.0501743046f);
    p = __builtin_fmaf(p, ax, 0.0889789874f);
    p = __builtin_fmaf(p, ax, -0.2145988016f);
    p = __builtin_fmaf(p, ax, 1.5707963050f);
    float r = fast_sqrtf(1.0f - ax) * p;     // acos(|x|)
    return (x < 0.0f) ? (PI_F - r) : r;
}

__global__ __launch_bounds__(256)
void kcl_pair_kernel(const float* __restrict__ X,
                     const float* __restrict__ Y,
                     const float* __restrict__ Wp,
                     const float* __restrict__ Wo,
                     float* __restrict__ partial)
{
    __shared__ float posA[IJ * POS_K];   // 8 KB
    __shared__ float posB[IJ * POS_K];   // 8 KB
    __shared__ float oriA[IJ * ORI_K];   // 12 KB
    __shared__ float oriB[IJ * ORI_K];   // 12 KB
    __shared__ float red[256];

    const int b    = blockIdx.x;
    const int tid  = threadIdx.x;
    const int lane = tid & 31;
    const int wave = tid >> 5;

    // ---------------- Stage A: pack features into LDS (one transform/thread) --
    {
        const int i = tid;
        const float4* xm = (const float4*)(X + ((size_t)b * IJ + i) * 16);
        const float4* ym = (const float4*)(Y + ((size_t)b * IJ + i) * 16);
        float4 xr0 = xm[0], xr1 = xm[1], xr2 = xm[2];
        float4 yr0 = ym[0], yr1 = ym[1], yr2 = ym[2];

        float nx = xr0.w*xr0.w + xr1.w*xr1.w + xr2.w*xr2.w;
        float ny = yr0.w*yr0.w + yr1.w*yr1.w + yr2.w*yr2.w;

        float* pa = &posA[i * POS_K];
        pa[0] = xr0.w; pa[1] = xr1.w; pa[2] = xr2.w;
        pa[3] = 1.0f;  pa[4] = nx;
        pa[5] = 0.0f;  pa[6] = 0.0f;  pa[7] = 0.0f;

        float* pb = &posB[i * POS_K];
        pb[0] = -2.0f*yr0.w; pb[1] = -2.0f*yr1.w; pb[2] = -2.0f*yr2.w;
        pb[3] = ny;   pb[4] = 1.0f;
        pb[5] = 0.0f; pb[6] = 0.0f; pb[7] = 0.0f;

        float* oa = &oriA[i * ORI_K];
        oa[0] = xr0.x; oa[1]  = xr0.y; oa[2]  = xr0.z;
        oa[3] = xr1.x; oa[4]  = xr1.y; oa[5]  = xr1.z;
        oa[6] = xr2.x; oa[7]  = xr2.y; oa[8]  = xr2.z;
        oa[9] = 0.0f;  oa[10] = 0.0f;  oa[11] = 0.0f;

        float* ob = &oriB[i * ORI_K];
        ob[0] = yr0.x; ob[1]  = yr0.y; ob[2]  = yr0.z;
        ob[3] = yr1.x; ob[4]  = yr1.y; ob[5]  = yr1.z;
        ob[6] = yr2.x; ob[7]  = yr2.y; ob[8]  = yr2.z;
        ob[9] = 0.0f;  ob[10] = 0.0f;  ob[11] = 0.0f;
    }

    // weights: C-layout -> VGPR index r == (i mod 8) == m; lane&7 == (j mod 8) == n
    float wp[8], wo[8];
    #pragma unroll
    for (int r = 0; r < 8; ++r) {
        wp[r] = Wp[r * 8 + (lane & 7)];
        wo[r] = Wo[r * 8 + (lane & 7)];
    }

    __syncthreads();

    // ---------------- Stage B: 16x16 tiles; each wave owns 2 tile-rows --------
    // A-frag (16x4 f32): lanes 0-15 -> K={0,1}, lanes 16-31 -> K={2,3};
    // VGPR picks within the pair. B mirrors with lane = N.
    const int l15 = lane & 15;
    const int kb  = (lane >> 4) << 1;   // 0 or 2

    float acc = 0.0f;
    #pragma unroll
    for (int rr = 0; rr < 2; ++rr) {                 // 16 tile rows / 8 waves
        const int ti = wave * 2 + rr;                // wave-uniform
        const int Mi = ti * 16 + l15;

        // Load A fragments once per tile-row, reuse across 16 column tiles.
        const float* pa = &posA[Mi * POS_K + kb];
        const float* oa = &oriA[Mi * ORI_K + kb];
        v2f aP0, aP1, aO0, aO1, aO2;
        aP0[0] = pa[0]; aP0[1] = pa[1];
        aP1[0] = pa[4]; aP1[1] = pa[5];
        aO0[0] = oa[0]; aO0[1] = oa[1];
        aO1[0] = oa[4]; aO1[1] = oa[5];
        aO2[0] = oa[8]; aO2[1] = oa[9];

        for (int tj = 0; tj < 16; ++tj) {            // wave-uniform: EXEC all-1s
            const int Nj = tj * 16 + l15;
            const float* pb = &posB[Nj * POS_K + kb];
            const float* ob = &oriB[Nj * ORI_K + kb];
            v2f bP0, bP1, bO0, bO1, bO2;
            bP0[0] = pb[0]; bP0[1] = pb[1];
            bP1[0] = pb[4]; bP1[1] = pb[5];
            bO0[0] = ob[0]; bO0[1] = ob[1];
            bO1[0] = ob[4]; bO1[1] = ob[5];
            bO2[0] = ob[8]; bO2[1] = ob[9];

            v8f cPos = {};
            cPos = __builtin_amdgcn_wmma_f32_16x16x4_f32(
                false, aP0, false, bP0, (short)0, cPos, false, false);
            cPos = __builtin_amdgcn_wmma_f32_16x16x4_f32(
                false, aP1, false, bP1, (short)0, cPos, false, false);

            v8f cOri = {};
            cOri = __builtin_amdgcn_wmma_f32_16x16x4_f32(
                false, aO0, false, bO0, (short)0, cOri, false, false);
            cOri = __builtin_amdgcn_wmma_f32_16x16x4_f32(
                false, aO1, false, bO1, (short)0, cOri, false, false);
            cOri = __builtin_amdgcn_wmma_f32_16x16x4_f32(
                false, aO2, false, bO2, (short)0, cOri, false, false);

            // Elementwise tail (VALU-bound part): fast sqrt + fast acos.
            #pragma unroll
            for (int r = 0; r < 8; ++r) {
                float d2 = cPos[r];
                float dp = fast_sqrtf(fmaxf(d2, 1e-30f));
                float c  = (cOri[r] - 1.0f) * (1.0f / (2.0f + EPSF));
                c = fminf(fmaxf(c, -1.0f + EPSF), 1.0f - EPSF);
                float dori = fast_acosf(c) * INV_PI;
                acc = __builtin_fmaf(dp,   wp[r], acc);
                acc = __builtin_fmaf(dori, wo[r], acc);
            }
        }
    }

    // ---------------- Block reduction (deterministic tree) --------------------
    red[tid] = acc;
    __syncthreads();
    #pragma unroll
    for (int s = 128; s > 0; s >>= 1) {
        if (tid < s) red[tid] += red[tid + s];
        __syncthreads();
    }
    if (tid == 0) partial[b] = red[0];
}

__global__ __launch_bounds__(256)
void kcl_reduce_kernel(const float* __restrict__ partial, int n,
                       float* __restrict__ out, float scale)
{
    __shared__ float red[256];
    const int tid = threadIdx.x;
    float v = 0.0f;
    for (int i = tid; i < n; i += 256) v += partial[i];
    red[tid] = v;
    __syncthreads();
    #pragma unroll
    for (int s = 128; s > 0; s >>= 1) {
        if (tid < s) red[tid] += red[tid + s];
        __syncthreads();
    }
    if (tid == 0) out[0] = red[0] * scale;
}

extern "C" void kernel_launch(void* const* d_in, const int* in_sizes, int n_in,
                              void* d_out, int out_size, void* d_ws, size_t ws_size,
                              hipStream_t stream) {
    const float* X  = (const float*)d_in[0];   // (B, S, M, 4, 4) f32
    const float* Y  = (const float*)d_in[1];   // (B, T, N, 4, 4) f32
    const float* Wp = (const float*)d_in[2];   // (8, 8) f32
    const float* Wo = (const float*)d_in[3];   // (8, 8) f32

    const int B = in_sizes[0] / (IJ * 16);     // 256 for the given shapes
    float* partial = (float*)d_ws;             // B floats of scratch

    kcl_pair_kernel<<<B, 256, 0, stream>>>(X, Y, Wp, Wo, partial);

    // mean over (B, S, T) = B * 32 * 32 elements
    const float scale = 1.0f / ((float)B * 32.0f * 32.0f);
    kcl_reduce_kernel<<<1, 256, 0, stream>>>(partial, B, (float*)d_out, scale);
}